// GNNLayer_12919261627019
// MI455X (gfx1250) — compile-verified
//
#include <hip/hip_runtime.h>

typedef float v2f __attribute__((ext_vector_type(2)));
typedef float v8f __attribute__((ext_vector_type(8)));
typedef float f4  __attribute__((ext_vector_type(4)));

#define NF        128   // feature dim F
#define KN        16    // neighbors per face
#define TWOF      256   // 2F (GEMM K)
#define OUTF      128   // output dim O
#define XS_STRIDE 260   // LDS row stride in floats: 4*M bank rotation -> conflict-free b64 reads
#define NEG_SLOPE 0.2f
#define LN_EPS    1e-5f

// One block (256 thr = 8 wave32) handles 16 rows. N = 100000 = 6250 * 16 (exact).
__global__ __launch_bounds__(256) void gnn_fused_kernel(
    const float* __restrict__ ff,     // [N, 128]
    const int*   __restrict__ adj,    // [N, 16]
    const float* __restrict__ W,      // [128, 256] (torch layout, h = x @ W^T)
    const float* __restrict__ bias,   // [128]
    const float* __restrict__ gamma,  // [128]
    const float* __restrict__ beta,   // [128]
    float*       __restrict__ out)    // [N, 128]
{
    __shared__ float x_s[16 * XS_STRIDE];   // concat tile [16][256] (padded)
    __shared__ int   adj_s[16 * KN];
    __shared__ float ps[8][16];             // per-wave partial row sums
    __shared__ float pq[8][16];             // per-wave partial row sumsq
    __shared__ float mu_s[16];
    __shared__ float rs_s[16];

    const int tid = threadIdx.x;
    const int r0  = blockIdx.x * 16;

    // ---- stage adjacency tile (16 rows x 16 neighbors) ----
    adj_s[tid] = adj[r0 * KN + tid];
    __syncthreads();

    // ---- neighbor-mean aggregation + self copy into LDS x-tile ----
    {
        const int r   = tid >> 4;        // row in tile
        const int sub = tid & 15;        // 8-float feature chunk
        const int c0  = sub * 8;
        f4 acc0 = {0.f, 0.f, 0.f, 0.f};
        f4 acc1 = {0.f, 0.f, 0.f, 0.f};
        int cnt = 0;
        #pragma unroll
        for (int k = 0; k < KN; ++k) {
            const int idx = adj_s[r * KN + k];
            if (idx >= 0) {                       // -1 = padding
                const f4* p = (const f4*)(ff + (long)idx * NF + c0);
                acc0 += p[0];
                acc1 += p[1];
                ++cnt;
            }
        }
        const float inv = 1.0f / (float)(cnt > 0 ? cnt : 1);
        const f4* pself = (const f4*)(ff + (long)(r0 + r) * NF + c0);
        f4 s0 = pself[0];
        f4 s1 = pself[1];
        float* xrow = &x_s[r * XS_STRIDE];
        *(f4*)(xrow + c0)          = s0;
        *(f4*)(xrow + c0 + 4)      = s1;
        *(f4*)(xrow + NF + c0)     = acc0 * inv;
        *(f4*)(xrow + NF + c0 + 4) = acc1 * inv;
    }
    __syncthreads();

    // ---- GEMM: wave w -> 16x16 output tile, cols [16w, 16w+16), K = 256 ----
    const int wave  = tid >> 5;
    const int lane  = tid & 31;
    const int lm    = lane & 15;
    const int half  = lane >> 4;
    const int o     = wave * 16 + lm;       // output column for this lane (B/C/D N index)

    // A (16x4 f32): lanes hold M = lane%16; VGPR0 K = 2*half, VGPR1 K = 2*half+1
    const float* aptr = &x_s[lm * XS_STRIDE + 2 * half];
    // B (4x16 f32): B[k][n] = W[o][k]; VGPR0 K = 2*half, VGPR1 K = 2*half+1
    const float* bptr = W + (long)o * TWOF + 2 * half;

    v8f acc = {0.f, 0.f, 0.f, 0.f, 0.f, 0.f, 0.f, 0.f};
    #pragma unroll 8
    for (int kk = 0; kk < TWOF / 4; ++kk) {
        v2f a = *(const v2f*)(aptr + 4 * kk);   // ds_load_b64, bank-conflict-free
        v2f b = *(const v2f*)(bptr + 4 * kk);   // global_load_b64, L2-resident W
        acc = __builtin_amdgcn_wmma_f32_16x16x4_f32(
            /*neg_a=*/false, a, /*neg_b=*/false, b,
            /*c_mod=*/(short)0, acc, /*reuse_a=*/false, /*reuse_b=*/false);
    }

    // ---- bias + per-row sum / sumsq (16-lane butterfly stays within wave halves) ----
    const float bv  = bias[o];
    const float gv  = gamma[o];
    const float btv = beta[o];
    float hv[8];
    #pragma unroll
    for (int g = 0; g < 8; ++g) {
        const float v = acc[g] + bv;   // D row M = g + 8*half, col N = lm
        hv[g] = v;
        float s = v;
        float q = v * v;
        #pragma unroll
        for (int m = 1; m <= 8; m <<= 1) {
            s += __shfl_xor(s, m, 32);
            q += __shfl_xor(q, m, 32);
        }
        if (lm == 0) {
            ps[wave][g + 8 * half] = s;
            pq[wave][g + 8 * half] = q;
        }
    }
    __syncthreads();

    // ---- cross-wave LayerNorm stats (one row per thread, tid < 16) ----
    if (tid < 16) {
        float s = 0.f, q = 0.f;
        #pragma unroll
        for (int w2 = 0; w2 < 8; ++w2) { s += ps[w2][tid]; q += pq[w2][tid]; }
        const float mu  = s * (1.0f / (float)OUTF);
        const float var = q * (1.0f / (float)OUTF) - mu * mu;
        mu_s[tid] = mu;
        rs_s[tid] = rsqrtf(var + LN_EPS);
    }
    __syncthreads();

    // ---- normalize, affine, LeakyReLU, store ----
    #pragma unroll
    for (int g = 0; g < 8; ++g) {
        const int   row = g + 8 * half;
        float v = (hv[g] - mu_s[row]) * rs_s[row] * gv + btv;
        v = (v >= 0.0f) ? v : NEG_SLOPE * v;
        out[(long)(r0 + row) * OUTF + o] = v;
    }
}

extern "C" void kernel_launch(void* const* d_in, const int* in_sizes, int n_in,
                              void* d_out, int out_size, void* d_ws, size_t ws_size,
                              hipStream_t stream) {
    const float* ff    = (const float*)d_in[0];   // face_features [N,128] f32
    const int*   adj   = (const int*)  d_in[1];   // adjacency     [N,16]  int
    const float* W     = (const float*)d_in[2];   // [128, 256]
    const float* b     = (const float*)d_in[3];   // [128]
    const float* gamma = (const float*)d_in[4];   // [128]
    const float* beta  = (const float*)d_in[5];   // [128]
    float*       out   = (float*)d_out;           // [N, 128]

    const int N = in_sizes[0] / NF;               // 100000, multiple of 16
    const int nblocks = N / 16;
    gnn_fused_kernel<<<nblocks, 256, 0, stream>>>(ff, adj, W, b, gamma, beta, out);
}